// _CustomMultiHeadAttention_46411416600765
// MI455X (gfx1250) — compile-verified
//
#include <hip/hip_runtime.h>
#include <cstdint>

#define SEQ 2048
#define NB 8
#define NH 5
#define HD 120

typedef float v2f  __attribute__((ext_vector_type(2)));
typedef float v4f  __attribute__((ext_vector_type(4)));
typedef float v8f  __attribute__((ext_vector_type(8)));
typedef unsigned int u32x4 __attribute__((ext_vector_type(4)));
typedef int   i32x4 __attribute__((ext_vector_type(4)));
typedef int   i32x8 __attribute__((ext_vector_type(8)));

__device__ __forceinline__ float fast_exp2(float x) {
  return __builtin_amdgcn_exp2f(x);
}

// ---------------------------------------------------------------------------
// Setup kernel 1: the 20 weight dot-products over d=120, via f32 WMMA.
//   ws[0..4]   = Wq_h . Wk_h          (dotQK)
//   ws[5..9]   = Bq_h . Wk_h          (dotBqWk)
//   ws[10..14] = wo_h . Wv_h          (P_h)
//   ws[16..20] = wo_h . Bv_h          (dotOBv)
// Two V_WMMA_F32_16X16X4_F32 chains, 30 K-steps each; diagonal = dots.
// ---------------------------------------------------------------------------
__global__ void setup_dots(const float* __restrict__ wq_w, const float* __restrict__ wq_b,
                           const float* __restrict__ wk_w,
                           const float* __restrict__ wv_w, const float* __restrict__ wv_b,
                           const float* __restrict__ wo_w,
                           float* __restrict__ ws) {
  const int lane = threadIdx.x;           // blockDim == 32
#if __has_builtin(__builtin_amdgcn_wmma_f32_16x16x4_f32)
  const int r  = lane & 15;               // A row == B col
  const int kh = lane >> 4;               // K half: 0 -> K{0,1}, 1 -> K{2,3}
  // chain 1: rows 0-4 Wq_h, 5-9 Bq_h, 10-14 wo_h ; cols 0-9 Wk_h, 10-14 Wv_h
  const float* L1 = (r < 5)  ? (wq_w + r * HD)
                  : (r < 10) ? (wq_b + (r - 5) * HD)
                  : (r < 15) ? (wo_w + (r - 10) * HD)
                             : wq_w;
  const float* R1 = (r < 10) ? (wk_w + (r % 5) * HD)
                  : (r < 15) ? (wv_w + (r - 10) * HD)
                             : wk_w;
  // chain 2: rows 0-4 wo_h ; cols 0-4 Bv_h
  const float* L2 = wo_w + ((r < 5) ? r * HD : 0);
  const float* R2 = wv_b + ((r < 5) ? r * HD : 0);

  v8f acc1 = {}; v8f acc2 = {};
  for (int k0 = 0; k0 < HD; k0 += 4) {
    const int k = k0 + 2 * kh;
    v2f a1; a1.x = L1[k]; a1.y = L1[k + 1];
    v2f b1; b1.x = R1[k]; b1.y = R1[k + 1];
    acc1 = __builtin_amdgcn_wmma_f32_16x16x4_f32(false, a1, false, b1, (short)0, acc1, false, false);
    v2f a2; a2.x = L2[k]; a2.y = L2[k + 1];
    v2f b2; b2.x = R2[k]; b2.y = R2[k + 1];
    acc2 = __builtin_amdgcn_wmma_f32_16x16x4_f32(false, a2, false, b2, (short)0, acc2, false, false);
  }
  // Diagonal (m,m): m<8 -> lane m, comp m ; m in 8..15 -> lane m+16, comp m-8.
  if (lane < 8)                    ws[lane]      = acc1[lane];
  if (lane >= 24 && lane <= 30)    ws[lane - 16] = acc1[lane - 24];
  if (lane < 5)                    ws[16 + lane] = acc2[lane];
#else
  if (lane < 20) {
    const int h = lane % 5;
    const float *a, *b;
    if (lane < 5)       { a = wq_w + h * HD; b = wk_w + h * HD; }
    else if (lane < 10) { a = wq_b + h * HD; b = wk_w + h * HD; }
    else if (lane < 15) { a = wo_w + h * HD; b = wv_w + h * HD; }
    else                { a = wo_w + h * HD; b = wv_b + h * HD; }
    float s = 0.f;
    for (int d = 0; d < HD; ++d) s += a[d] * b[d];
    ws[(lane < 15) ? lane : (16 + h)] = s;
  }
#endif
}

// ---------------------------------------------------------------------------
// Setup kernel 2: per-batch max/min of x   ->  ws[32+b], ws[40+b]
// ---------------------------------------------------------------------------
__global__ void setup_minmax(const float* __restrict__ x, float* __restrict__ ws) {
  const int wid  = threadIdx.x >> 5;      // batch (blockDim 256 = 8 waves)
  const int lane = threadIdx.x & 31;
  const float* xb = x + wid * SEQ;
  float mx = -3.4e38f, mn = 3.4e38f;
  for (int j = lane; j < SEQ; j += 32) {
    float v = xb[j];
    mx = fmaxf(mx, v); mn = fminf(mn, v);
  }
#pragma unroll
  for (int off = 16; off; off >>= 1) {
    mx = fmaxf(mx, __shfl_xor(mx, off, 32));
    mn = fminf(mn, __shfl_xor(mn, off, 32));
  }
  if (lane == 0) { ws[32 + wid] = mx; ws[40 + wid] = mn; }
}

// ---------------------------------------------------------------------------
// Main kernel: one wave32 per query row i.  2048 blocks x 256 threads.
//   pass 1: per head, s1 = sum_j exp2(a*x_j - t), s2 = sum_j x_j*exp2(...)
//   pass 2: stream avg_attn row with float4 nontemporal stores.
// x row staged in LDS via TDM tensor_load_to_lds (CDNA5 async-tensor path).
// ---------------------------------------------------------------------------
__global__ __launch_bounds__(256) void attn_main(const float* __restrict__ x,
                                                 const float* __restrict__ wo_b,
                                                 const float* __restrict__ ws,
                                                 float* __restrict__ out,
                                                 float* __restrict__ avg) {
  __shared__ float lx[SEQ];
  const int batch = blockIdx.x >> 8;      // SEQ/8 = 256 tiles per batch
  const int tile  = blockIdx.x & 255;
  const int tid   = threadIdx.x;
  const int lane  = tid & 31;
  const int wid   = tid >> 5;
  const float* xb = x + batch * SEQ;
  __builtin_prefetch(xb, 0, 0);           // global_prefetch_b8

#if __has_builtin(__builtin_amdgcn_tensor_load_to_lds) && __has_builtin(__builtin_amdgcn_s_wait_tensorcnt)
  if (tid == 0) {
    const unsigned long long ga = (unsigned long long)(uintptr_t)xb;
    u32x4 g0;
    g0.x = 1u;                                            // count=1, user desc
    g0.y = (unsigned int)(uintptr_t)&lx[0];               // lds_addr
    g0.z = (unsigned int)(ga & 0xffffffffu);              // global_addr[31:0]
    g0.w = (unsigned int)((ga >> 32) & 0x1ffffffu) | (2u << 30); // addr[56:32], type=2
    i32x8 g1;
    g1[0] = (2 << 16);                                    // data_size = 4B
    g1[1] = (int)((SEQ & 0xffff) << 16);                  // tensor_dim0[15:0]
    g1[2] = (int)((SEQ >> 16) | (1 << 16));               // dim0 hi, tensor_dim1 = 1
    g1[3] = (int)(SEQ << 16);                             // tile_dim0 = 2048
    g1[4] = 1;                                            // tile_dim1 = 1
    g1[5] = SEQ;                                          // tensor_dim0_stride
    g1[6] = 0;
    g1[7] = 0;
    i32x4 z4 = {0, 0, 0, 0};
    i32x8 z8 = {0, 0, 0, 0, 0, 0, 0, 0};
    __builtin_amdgcn_tensor_load_to_lds(g0, g1, z4, z4, z8, 0);
    __builtin_amdgcn_s_wait_tensorcnt(0);
  }
#else
  for (int j = tid; j < SEQ; j += 256) lx[j] = xb[j];
#endif
  __syncthreads();

  const int   i  = tile * 8 + wid;
  const float xi = lx[i];
  const float mxb = ws[32 + batch], mnb = ws[40 + batch];
  const float SCALE = 0.0912870929175277f;     // 120^-0.5
  const float LOG2E = 1.4426950408889634f;

  float a2[NH], t[NH], P[NH];
  float o = wo_b[0];
#pragma unroll
  for (int h = 0; h < NH; ++h) {
    const float alpha = SCALE * fmaf(xi, ws[h], ws[5 + h]);  // x_j coefficient
    a2[h] = alpha * LOG2E;                                   // base-2 domain
    t[h]  = (a2[h] >= 0.f) ? a2[h] * mxb : a2[h] * mnb;      // row max (stable)
    P[h]  = ws[10 + h];
    o    += ws[16 + h];                                      // wo . Bv_h
  }

  // ---- pass 1: denominators and weighted sums -----------------------------
  float s1[NH] = {0, 0, 0, 0, 0}, s2[NH] = {0, 0, 0, 0, 0};
  for (int j = lane; j < SEQ; j += 32) {
    const float xv = lx[j];
#pragma unroll
    for (int h = 0; h < NH; ++h) {
      const float e = fast_exp2(fmaf(a2[h], xv, -t[h]));
      s1[h] += e;
      s2[h]  = fmaf(xv, e, s2[h]);
    }
  }
#pragma unroll
  for (int off = 16; off; off >>= 1) {
#pragma unroll
    for (int h = 0; h < NH; ++h) {
      s1[h] += __shfl_xor(s1[h], off, 32);
      s2[h] += __shfl_xor(s2[h], off, 32);
    }
  }

  float c[NH];
#pragma unroll
  for (int h = 0; h < NH; ++h) {
    const float inv = 1.f / s1[h];
    o    = fmaf(P[h], s2[h] * inv, o);     // P_h * m_h
    c[h] = 0.2f * inv;                     // (1/H)/denom
  }
  if (lane == 0) out[batch * SEQ + i] = o;

  // ---- pass 2: stream the head-averaged attention row ---------------------
  float* __restrict__ row = avg + ((size_t)batch * SEQ + (size_t)i) * SEQ;
  for (int j = lane * 4; j < SEQ; j += 128) {
    const v4f xv = *(const v4f*)&lx[j];
    v4f r;
#pragma unroll
    for (int e4 = 0; e4 < 4; ++e4) {
      const float xval = xv[e4];
      float acc = 0.f;
#pragma unroll
      for (int h = 0; h < NH; ++h)
        acc = fmaf(c[h], fast_exp2(fmaf(a2[h], xval, -t[h])), acc);
      r[e4] = acc;
    }
    __builtin_nontemporal_store(r, (v4f*)&row[j]);
  }
}

// ---------------------------------------------------------------------------
extern "C" void kernel_launch(void* const* d_in, const int* in_sizes, int n_in,
                              void* d_out, int out_size, void* d_ws, size_t ws_size,
                              hipStream_t stream) {
  const float* x    = (const float*)d_in[0];
  const float* wq_w = (const float*)d_in[1];
  const float* wq_b = (const float*)d_in[2];
  const float* wk_w = (const float*)d_in[3];
  // d_in[4] = wk_b : cancels inside softmax, unused.
  const float* wv_w = (const float*)d_in[5];
  const float* wv_b = (const float*)d_in[6];
  const float* wo_w = (const float*)d_in[7];
  const float* wo_b = (const float*)d_in[8];

  float* outp = (float*)d_out;                 // (B,S,1) = 16384 floats
  float* avgp = outp + NB * SEQ;               // (B,S,S) follows
  float* wsf  = (float*)d_ws;

  setup_dots<<<1, 32, 0, stream>>>(wq_w, wq_b, wk_w, wv_w, wv_b, wo_w, wsf);
  setup_minmax<<<1, 256, 0, stream>>>(x, wsf);
  attn_main<<<NB * (SEQ / 8), 256, 0, stream>>>(x, wo_b, wsf, outp, avgp);
}